// AttentionRNN_59433757442512
// MI455X (gfx1250) — compile-verified
//
#include <hip/hip_runtime.h>
#include <math.h>
#include <stdint.h>

// ---------------- problem constants (from reference) ----------------
#define BB 32
#define PP 512
#define QQ 64
#define EE 300
#define HH2 256      // H
#define HHH 128      // HH = H/2
#define G3 384       // 3*HH
#define NEGV (-1e7f)

typedef __attribute__((ext_vector_type(2))) float v2f;
typedef __attribute__((ext_vector_type(8))) float v8f;

// fp32 WMMA 16x16 tile, K step 4.  D = A(16x4) * B(4x16) + C
__device__ __forceinline__ v8f wmma4(v2f a, v2f b, v8f c) {
  return __builtin_amdgcn_wmma_f32_16x16x4_f32(
      /*neg_a=*/false, a, /*neg_b=*/false, b,
      /*c_mod=*/(short)0, c, /*reuse_a=*/false, /*reuse_b=*/false);
}

// =====================================================================
// K1: xp[m, n] = emb[tok[m]] . wih[n, :] + bih[n]   (M = B*T, N = 384, K = 300)
// one wave per 16x64 strip (4 N-tiles): A fragment reused 4x per K-step.
// =====================================================================
__global__ void __launch_bounds__(32)
embed_proj_kernel(const int* __restrict__ tok, const float* __restrict__ emb,
                  const float* __restrict__ wih, const float* __restrict__ bih,
                  float* __restrict__ xp) {
  const int lane = threadIdx.x;
  const int hi = lane >> 4, l = lane & 15;
  const int mt = blockIdx.x, ng = blockIdx.y;   // ng indexes a group of 4 N tiles
  const int row = mt * 16 + l;                  // A-fragment row for this lane
  const int t = tok[row];
  const float* __restrict__ arow = emb + (long)t * EE;

  const float* brow[4]; int ncol[4];
  v8f acc[4];
#pragma unroll
  for (int j = 0; j < 4; ++j) {
    ncol[j] = ng * 64 + j * 16 + l;
    brow[j] = wih + (long)ncol[j] * EE;
    const float bv = bih[ncol[j]];
#pragma unroll
    for (int r = 0; r < 8; ++r) acc[j][r] = bv;
  }

  for (int k0 = 0; k0 < EE; k0 += 4) {          // 300/4 = 75 exact
    const int k = k0 + 2 * hi;
    v2f a = *(const v2f*)(arow + k);
#pragma unroll
    for (int j = 0; j < 4; ++j) {
      v2f b = *(const v2f*)(brow[j] + k);
      acc[j] = wmma4(a, b, acc[j]);
    }
  }
#pragma unroll
  for (int j = 0; j < 4; ++j)
#pragma unroll
    for (int r = 0; r < 8; ++r) {
      const int m = mt * 16 + r + 8 * hi;
      xp[(long)m * G3 + ncol[j]] = acc[j][r];
    }
}

// =====================================================================
// K2: persistent GRU scan.  One workgroup (16 waves) per (sequence,direction).
// Whh (384x128) loaded to LDS through the ASYNCcnt path (async-to-LDS B64),
// bank-padded to stride 130.  h (32x128) and hp (32x384) live in LDS.
// Per step: gate inputs prefetched to registers, then 48 WMMA tiles -> hp,
// then 512-thread gate phase.
// =====================================================================
struct GruCfg {
  const float* xp;    // (B,T,384) precomputed x@Wih^T + bih
  const float* whh;   // (384,128)
  const float* bhh;   // (384,)
  const int*   tok;   // (B,T)
  float*       out;   // (B,T,H) half written at dirOff
  int T, dirOff, rev;
};
struct GruCfg4 { GruCfg c[4]; };

#define WPAD 130
#define HPAD 130

__global__ void __launch_bounds__(512)
gru_scan_kernel(GruCfg4 cfgs) {
  const GruCfg cfg = cfgs.c[blockIdx.x];
  extern __shared__ float lds[];
  float* whh_s = lds;                       // 384*130
  float* h_s   = whh_s + G3 * WPAD;         // 32*130
  float* hp_s  = h_s + BB * HPAD;           // 32*384
  float* bhh_s = hp_s + BB * G3;            // 384

  const int tid = threadIdx.x;

  // ---- async preload of recurrent weights into padded LDS (B64 = 2 floats) ----
  for (int i = tid; i < G3 * HHH / 2; i += 512) {
    const int n = i >> 6;                   // 64 float-pairs per 128-wide row
    const int k = (i & 63) * 2;
    const unsigned lds_off =
        (unsigned)(uintptr_t)(whh_s + n * WPAD + k);   // low 32b = LDS byte addr
    const unsigned long long ga =
        (unsigned long long)(uintptr_t)(cfg.whh + n * HHH + k);
    asm volatile("global_load_async_to_lds_b64 %0, %1, off"
                 :: "v"(lds_off), "v"(ga) : "memory");
  }
  if (tid < G3) bhh_s[tid] = cfg.bhh[tid];
  for (int i = tid; i < BB * HHH; i += 512) {
    const int b = i >> 7, j = i & 127;
    h_s[b * HPAD + j] = 0.f;
  }
  asm volatile("s_wait_asynccnt 0" ::: "memory");
  __syncthreads();

  const int wave = tid >> 5, lane = tid & 31;
  const int hi = lane >> 4, l = lane & 15;

  for (int s = 0; s < cfg.T; ++s) {
    const int t = cfg.rev ? (cfg.T - 1 - s) : s;

    // ---- prefetch this step's gate inputs into registers; these global
    //      loads overlap the LDS-fed WMMA phase below ----
    float pxr[8], pxz[8], pxn[8]; int ptk[8];
#pragma unroll
    for (int i = 0; i < 8; ++i) {
      const int e = tid + i * 512;
      const int b = e >> 7, j = e & 127;
      const float* xrow = cfg.xp + ((long)b * cfg.T + t) * G3;
      pxr[i] = xrow[j];
      pxz[i] = xrow[HHH + j];
      pxn[i] = xrow[2 * HHH + j];
      ptk[i] = cfg.tok[b * cfg.T + t];
    }

    // ---- hp = h @ Whh^T + bhh : 48 tiles over 16 waves (3 each) ----
#pragma unroll
    for (int i = 0; i < 3; ++i) {
      const int tile = wave + 16 * i;      // 0..47
      const int mt = tile & 1, nt = tile >> 1;
      const int m = mt * 16 + l;
      const int n = nt * 16 + l;
      v8f acc;
      const float bv = bhh_s[n];
#pragma unroll
      for (int r = 0; r < 8; ++r) acc[r] = bv;
      for (int k0 = 0; k0 < HHH; k0 += 4) {
        const int k = k0 + 2 * hi;
        v2f a = *(const v2f*)(h_s + m * HPAD + k);
        v2f b = *(const v2f*)(whh_s + n * WPAD + k);
        acc = wmma4(a, b, acc);
      }
#pragma unroll
      for (int r = 0; r < 8; ++r)
        hp_s[(mt * 16 + r + 8 * hi) * G3 + n] = acc[r];
    }
    __syncthreads();

    // ---- gates: 4096 h-elements over 512 threads ----
#pragma unroll
    for (int i = 0; i < 8; ++i) {
      const int e = tid + i * 512;
      const int b = e >> 7, j = e & 127;
      const float hr = hp_s[b * G3 + j];
      const float hz = hp_s[b * G3 + HHH + j];
      const float hn = hp_s[b * G3 + 2 * HHH + j];
      const float hprev = h_s[b * HPAD + j];
      const float r = 1.f / (1.f + __expf(-(pxr[i] + hr)));
      const float z = 1.f / (1.f + __expf(-(pxz[i] + hz)));
      const float nn = tanhf(pxn[i] + r * hn);
      const float hnew = (1.f - z) * nn + z * hprev;
      const bool msk = ptk[i] != 0;
      h_s[b * HPAD + j] = msk ? hnew : hprev;
      cfg.out[((long)b * cfg.T + t) * HH2 + cfg.dirOff + j] = msk ? hnew : 0.f;
    }
    __syncthreads();
  }
}

// =====================================================================
// K3: pw[b,p] = penc . w1 ; qw[b,q] = qenc . w2   (one wave per row)
// =====================================================================
__global__ void __launch_bounds__(32)
rowdot_kernel(const float* __restrict__ penc, const float* __restrict__ qenc,
              const float* __restrict__ attn_w, float* __restrict__ pw,
              float* __restrict__ qw) {
  const int row = blockIdx.x, lane = threadIdx.x;
  const float* vec; const float* w; float* out; int orow;
  if (row < BB * PP) { vec = penc + (long)row * HH2; w = attn_w;        out = pw; orow = row; }
  else { orow = row - BB * PP; vec = qenc + (long)orow * HH2; w = attn_w + HH2; out = qw; }
  float s = 0.f;
  for (int h = lane; h < HH2; h += 32) s += vec[h] * w[h];
  for (int off = 16; off > 0; off >>= 1) s += __shfl_down(s, off, 32);
  if (lane == 0) out[orow] = s;
}

// =====================================================================
// K4: logits[b,p,q] = (penc*w3).qenc + pw[p] + qw[q] + attn_b, Q-masked.
// one wave per 16x64 strip (whole Q extent), K = H = 256.
// =====================================================================
__global__ void __launch_bounds__(32)
attn_logits_kernel(const float* __restrict__ penc, const float* __restrict__ qenc,
                   const float* __restrict__ attn_w, const float* __restrict__ attn_b,
                   const float* __restrict__ pw, const float* __restrict__ qw,
                   const int* __restrict__ qtok, float* __restrict__ logits) {
  const int lane = threadIdx.x, hi = lane >> 4, l = lane & 15;
  const int mt = blockIdx.x, b = blockIdx.y;
  const int m = mt * 16 + l;             // p row
  const float* __restrict__ arow = penc + ((long)b * PP + m) * HH2;
  const float* __restrict__ w3 = attn_w + 2 * HH2;
  const float* brow[4];
#pragma unroll
  for (int j = 0; j < 4; ++j)
    brow[j] = qenc + ((long)b * QQ + j * 16 + l) * HH2;

  v8f acc[4] = {};
  for (int k0 = 0; k0 < HH2; k0 += 4) {
    const int k = k0 + 2 * hi;
    v2f a = *(const v2f*)(arow + k);
    v2f w = *(const v2f*)(w3 + k);
    a.x *= w.x; a.y *= w.y;
#pragma unroll
    for (int j = 0; j < 4; ++j) {
      v2f bfrag = *(const v2f*)(brow[j] + k);
      acc[j] = wmma4(a, bfrag, acc[j]);
    }
  }
  const float ab = attn_b[0];
#pragma unroll
  for (int j = 0; j < 4; ++j) {
    const int n = j * 16 + l;            // q col
    const float qv = qw[b * QQ + n];
    const bool qmask = qtok[b * QQ + n] != 0;
#pragma unroll
    for (int r = 0; r < 8; ++r) {
      const int mr = mt * 16 + r + 8 * hi;
      float v = acc[j][r] + pw[b * PP + mr] + qv + ab;
      if (!qmask) v = NEGV;
      logits[((long)b * PP + mr) * QQ + n] = v;
    }
  }
}

// =====================================================================
// K5: in-place softmax over Q=64 (one wave per (b,p) row; 2 elems/lane)
// =====================================================================
__global__ void __launch_bounds__(32)
softmaxq_kernel(float* __restrict__ probs) {
  const int lane = threadIdx.x;
  float* row = probs + (long)blockIdx.x * QQ;
  float v0 = row[lane], v1 = row[lane + 32];
  float m = fmaxf(v0, v1);
  for (int off = 16; off > 0; off >>= 1) m = fmaxf(m, __shfl_xor(m, off, 32));
  const float e0 = __expf(v0 - m), e1 = __expf(v1 - m);
  float s = e0 + e1;
  for (int off = 16; off > 0; off >>= 1) s += __shfl_xor(s, off, 32);
  const float inv = 1.f / s;
  row[lane] = e0 * inv;
  row[lane + 32] = e1 * inv;
}

// =====================================================================
// K6: attw[b,p,h] = probs[b,p,:] @ qenc[b,:,h]   (M=512, N=256, K=64)
// one wave per 16x64 strip (4 N-tiles), A fragment reused 4x.
// =====================================================================
__global__ void __launch_bounds__(32)
attw_kernel(const float* __restrict__ probs, const float* __restrict__ qenc,
            float* __restrict__ attw) {
  const int lane = threadIdx.x, hi = lane >> 4, l = lane & 15;
  const int mt = blockIdx.x, ng = blockIdx.y, b = blockIdx.z;
  const int m = mt * 16 + l;
  const float* __restrict__ arow = probs + ((long)b * PP + m) * QQ;
  v8f acc[4] = {};
  for (int k0 = 0; k0 < QQ; k0 += 4) {
    const int k = k0 + 2 * hi;
    v2f a = *(const v2f*)(arow + k);
    const float* q0 = qenc + ((long)b * QQ + k) * HH2;
    const float* q1 = qenc + ((long)b * QQ + k + 1) * HH2;
#pragma unroll
    for (int j = 0; j < 4; ++j) {
      const int n = ng * 64 + j * 16 + l;
      v2f bfrag;
      bfrag.x = q0[n];
      bfrag.y = q1[n];
      acc[j] = wmma4(a, bfrag, acc[j]);
    }
  }
#pragma unroll
  for (int j = 0; j < 4; ++j) {
    const int n = ng * 64 + j * 16 + l;
#pragma unroll
    for (int r = 0; r < 8; ++r) {
      const int mr = mt * 16 + r + 8 * hi;
      attw[((long)b * PP + mr) * HH2 + n] = acc[j][r];
    }
  }
}

// =====================================================================
// K7: start/end = [penc; attw; penc*attw] . w + b, P-masked (wave per row)
// =====================================================================
__global__ void __launch_bounds__(32)
heads_kernel(const float* __restrict__ penc, const float* __restrict__ attw,
             const float* __restrict__ sw, const float* __restrict__ sb,
             const float* __restrict__ ew, const float* __restrict__ eb,
             const int* __restrict__ ptok, float* __restrict__ out_start,
             float* __restrict__ out_end) {
  const int row = blockIdx.x, lane = threadIdx.x;
  const float* pe = penc + (long)row * HH2;
  const float* aw = attw + (long)row * HH2;
  float ss = 0.f, se = 0.f;
  for (int h = lane; h < HH2; h += 32) {
    const float x = pe[h], y = aw[h], xy = x * y;
    ss += x * sw[h] + y * sw[HH2 + h] + xy * sw[2 * HH2 + h];
    se += x * ew[h] + y * ew[HH2 + h] + xy * ew[2 * HH2 + h];
  }
  for (int off = 16; off > 0; off >>= 1) {
    ss += __shfl_down(ss, off, 32);
    se += __shfl_down(se, off, 32);
  }
  if (lane == 0) {
    const bool msk = ptok[row] != 0;
    out_start[row] = msk ? ss + sb[0] : NEGV;
    out_end[row]   = msk ? se + eb[0] : NEGV;
  }
}

// =====================================================================
// K8: log_softmax over P=512 (one 512-thread block per batch row)
// =====================================================================
__global__ void __launch_bounds__(512)
logsoftmax_kernel(const float* __restrict__ in, float* __restrict__ out) {
  __shared__ float sm[17];
  const int tid = threadIdx.x, lane = tid & 31, wave = tid >> 5;
  const float v = in[(long)blockIdx.x * PP + tid];
  float m = v;
  for (int off = 16; off > 0; off >>= 1) m = fmaxf(m, __shfl_xor(m, off, 32));
  if (lane == 0) sm[wave] = m;
  __syncthreads();
  if (tid == 0) {
    float t = sm[0];
    for (int i = 1; i < 16; ++i) t = fmaxf(t, sm[i]);
    sm[16] = t;
  }
  __syncthreads();
  const float mx = sm[16];
  float s = __expf(v - mx);
  for (int off = 16; off > 0; off >>= 1) s += __shfl_xor(s, off, 32);
  __syncthreads();
  if (lane == 0) sm[wave] = s;
  __syncthreads();
  if (tid == 0) {
    float t = 0.f;
    for (int i = 0; i < 16; ++i) t += sm[i];
    sm[16] = logf(t);
  }
  __syncthreads();
  out[(long)blockIdx.x * PP + tid] = v - mx - sm[16];
}

// =====================================================================
extern "C" void kernel_launch(void* const* d_in, const int* in_sizes, int n_in,
                              void* d_out, int out_size, void* d_ws, size_t ws_size,
                              hipStream_t stream) {
  (void)in_sizes; (void)n_in; (void)out_size; (void)ws_size;
  const int*   passage  = (const int*)d_in[0];
  const int*   question = (const int*)d_in[1];
  const float* emb      = (const float*)d_in[2];
  const float* p_wih_f = (const float*)d_in[3],  *p_whh_f = (const float*)d_in[4];
  const float* p_bih_f = (const float*)d_in[5],  *p_bhh_f = (const float*)d_in[6];
  const float* p_wih_b = (const float*)d_in[7],  *p_whh_b = (const float*)d_in[8];
  const float* p_bih_b = (const float*)d_in[9],  *p_bhh_b = (const float*)d_in[10];
  const float* q_wih_f = (const float*)d_in[11], *q_whh_f = (const float*)d_in[12];
  const float* q_bih_f = (const float*)d_in[13], *q_bhh_f = (const float*)d_in[14];
  const float* q_wih_b = (const float*)d_in[15], *q_whh_b = (const float*)d_in[16];
  const float* q_bih_b = (const float*)d_in[17], *q_bhh_b = (const float*)d_in[18];
  const float* attn_w  = (const float*)d_in[19], *attn_b  = (const float*)d_in[20];
  const float* start_w = (const float*)d_in[21], *start_b = (const float*)d_in[22];
  const float* end_w   = (const float*)d_in[23], *end_b   = (const float*)d_in[24];

  // workspace layout (floats)
  float* ws = (float*)d_ws;
  float* xp_pf = ws;                                   // B*P*384
  float* xp_pb = xp_pf + (size_t)BB * PP * G3;
  float* xp_qf = xp_pb + (size_t)BB * PP * G3;         // B*Q*384
  float* xp_qb = xp_qf + (size_t)BB * QQ * G3;
  float* penc  = xp_qb + (size_t)BB * QQ * G3;         // B*P*H
  float* qenc  = penc + (size_t)BB * PP * HH2;         // B*Q*H
  float* probs = qenc + (size_t)BB * QQ * HH2;         // B*P*Q (logits -> probs)
  float* pw    = probs + (size_t)BB * PP * QQ;         // B*P
  float* qw    = pw + (size_t)BB * PP;                 // B*Q
  float* attw  = qw + (size_t)BB * QQ;                 // B*P*H

  float* out_start = (float*)d_out;
  float* out_end   = out_start + (size_t)BB * PP;
  float* out_ls    = out_end + (size_t)BB * PP;
  float* out_le    = out_ls + (size_t)BB * PP;

  // ---- K1: embedding gather + input projections (4 GEMMs) ----
  embed_proj_kernel<<<dim3(BB * PP / 16, G3 / 64), 32, 0, stream>>>(passage,  emb, p_wih_f, p_bih_f, xp_pf);
  embed_proj_kernel<<<dim3(BB * PP / 16, G3 / 64), 32, 0, stream>>>(passage,  emb, p_wih_b, p_bih_b, xp_pb);
  embed_proj_kernel<<<dim3(BB * QQ / 16, G3 / 64), 32, 0, stream>>>(question, emb, q_wih_f, q_bih_f, xp_qf);
  embed_proj_kernel<<<dim3(BB * QQ / 16, G3 / 64), 32, 0, stream>>>(question, emb, q_wih_b, q_bih_b, xp_qb);

  // ---- K2: 4 parallel persistent GRU scans ----
  GruCfg4 cfgs;
  cfgs.c[0] = { xp_pf, p_whh_f, p_bhh_f, passage,  penc, PP, 0,   0 };
  cfgs.c[1] = { xp_pb, p_whh_b, p_bhh_b, passage,  penc, PP, HHH, 1 };
  cfgs.c[2] = { xp_qf, q_whh_f, q_bhh_f, question, qenc, QQ, 0,   0 };
  cfgs.c[3] = { xp_qb, q_whh_b, q_bhh_b, question, qenc, QQ, HHH, 1 };
  const size_t ldsBytes = (size_t)(G3 * WPAD + BB * HPAD + BB * G3 + G3) * sizeof(float);
  (void)hipFuncSetAttribute(reinterpret_cast<const void*>(gru_scan_kernel),
                            hipFuncAttributeMaxDynamicSharedMemorySize, (int)ldsBytes);
  gru_scan_kernel<<<4, 512, ldsBytes, stream>>>(cfgs);

  // ---- K3: rank-1 attention terms ----
  rowdot_kernel<<<BB * PP + BB * QQ, 32, 0, stream>>>(penc, qenc, attn_w, pw, qw);

  // ---- K4: attention logits (full Q strip per wave) ----
  attn_logits_kernel<<<dim3(PP / 16, BB), 32, 0, stream>>>(
      penc, qenc, attn_w, attn_b, pw, qw, question, probs);

  // ---- K5: masked softmax over Q ----
  softmaxq_kernel<<<BB * PP, 32, 0, stream>>>(probs);

  // ---- K6: attended question vectors ----
  attw_kernel<<<dim3(PP / 16, HH2 / 64, BB), 32, 0, stream>>>(probs, qenc, attw);

  // ---- K7: start/end heads ----
  heads_kernel<<<BB * PP, 32, 0, stream>>>(penc, attw, start_w, start_b,
                                           end_w, end_b, passage, out_start, out_end);

  // ---- K8: log-softmax over P ----
  logsoftmax_kernel<<<BB, 512, 0, stream>>>(out_start, out_ls);
  logsoftmax_kernel<<<BB, 512, 0, stream>>>(out_end, out_le);
}